// DTWLoss_10144712753758
// MI455X (gfx1250) — compile-verified
//
#include <hip/hip_runtime.h>
#include <math.h>

// DTW DP, 2048x2048, squared-difference cost, output = sqrt(DTW[last,last]).
// Latency-bound wavefront DP: one wave32, 32 rows per lane, 2 strips.
// CDNA5 paths: async global->LDS staging (ASYNCcnt) + ds_bpermute shuffles.

#define LEN     2048
#define LANES   32
#define BROWS   32                       // rows per lane per strip
#define NSTRIPS (LEN / (LANES * BROWS))  // 2

#if __has_builtin(__builtin_amdgcn_global_load_async_to_lds_b128) && \
    __has_builtin(__builtin_amdgcn_s_wait_asynccnt)
#define DTW_ASYNC_LDS 1
typedef int dtw_v4i __attribute__((vector_size(4 * sizeof(int))));
#endif

__global__ __launch_bounds__(LANES)
void dtw_wavefront_kernel(const float* __restrict__ x,
                          const float* __restrict__ y,
                          float* __restrict__ out)
{
    __shared__ float sy[LEN];      // y staged in LDS (conflict-free reads)
    __shared__ float sBound[LEN];  // bottom row of strip 0 -> top input of strip 1

    const int   lane = (int)threadIdx.x;
    const float INF  = __builtin_inff();

    // ---- stage y[0..2047] into LDS --------------------------------------
#ifdef DTW_ASYNC_LDS
    // 2048 f32 = 512 x B128; 32 lanes x 16 iterations, per-lane addresses.
    #pragma unroll
    for (int i = 0; i < LEN / (LANES * 4); ++i) {
        const int e = (i * LANES + lane) * 4;   // float index, 16B aligned
        __builtin_amdgcn_global_load_async_to_lds_b128(
            (dtw_v4i*)(y + e),   // per-lane global source
            (dtw_v4i*)(sy + e),  // per-lane LDS destination
            0, 0);               // imm offset, cpol
    }
    __builtin_amdgcn_s_wait_asynccnt(0);
#else
    for (int i = lane; i < LEN; i += LANES) sy[i] = y[i];
#endif
    __syncthreads();  // single-wave workgroup: lowers to S_NOP, acts as fence

    float res = 0.0f;

    for (int strip = 0; strip < NSTRIPS; ++strip) {
        // Lane r owns global rows strip*1024 + r*32 .. +31 (contiguous).
        float xr[BROWS];  // x value for each owned row
        float L[BROWS];   // "left" value (previous column) per owned row
        #pragma unroll
        for (int k = 0; k < BROWS; ++k) {
            xr[k] = x[strip * (LANES * BROWS) + lane * BROWS + k];
            L[k]  = INF;  // left of column 0 is +inf
        }
        float diagTop = INF;  // diag input for this lane's top row
        float vbot    = INF;  // this lane's bottom-row value (shuffled down)

        for (int t = 0; t < LEN + LANES - 1; ++t) {
            // up for our top row = bottom row of lane-1, produced last step
            const float sh = __shfl_up(vbot, 1);

            const int  j      = t - lane;                     // our column
            const bool active = ((unsigned)j < (unsigned)LEN);
            const int  jc     = active ? j : 0;

            const float yj = sy[jc];
            const float bv = sBound[jc];  // garbage in strip 0, selected away

            float upTop = (lane == 0) ? ((strip == 0) ? INF : bv) : sh;
            // Cell (0,0): min-term must be 0 so v = cost  (row 0 -> cumsum)
            if (strip == 0 && lane == 0 && t == 0) upTop = 0.0f;

            float vprev = upTop;    // "up" input for row k
            float dprev = diagTop;  // "diag" input for row k
            #pragma unroll
            for (int k = 0; k < BROWS; ++k) {
                float c = xr[k] - yj;
                c *= c;                                        // cost(i,j)
                const float v = c + fminf(vprev, fminf(L[k], dprev));
                dprev = L[k];              // old left becomes next row's diag
                L[k]  = active ? v : L[k]; // commit left only when in-range
                vprev = v;                 // becomes "up" for row k+1
            }

            if (active) {
                diagTop = upTop;   // this step's up is next step's diag
                vbot    = vprev;   // bottom-row value for the lane below
                if (strip == 0 && lane == LANES - 1)
                    sBound[j] = vprev;                  // boundary row 1023
                if (strip == NSTRIPS - 1 && lane == LANES - 1 && j == LEN - 1)
                    res = vprev;                        // DTW[2047][2047]
            }
        }
    }

    if (lane == LANES - 1) out[0] = sqrtf(res);
}

extern "C" void kernel_launch(void* const* d_in, const int* in_sizes, int n_in,
                              void* d_out, int out_size, void* d_ws, size_t ws_size,
                              hipStream_t stream)
{
    (void)in_sizes; (void)n_in; (void)out_size; (void)d_ws; (void)ws_size;
    const float* x = (const float*)d_in[0];
    const float* y = (const float*)d_in[1];
    float* out = (float*)d_out;
    dtw_wavefront_kernel<<<1, LANES, 0, stream>>>(x, y, out);
}